// WorldModel_32882269618756
// MI455X (gfx1250) — compile-verified
//
#include <hip/hip_runtime.h>
#include <cstdint>

// ---------------------------------------------------------------------------
// WorldModel step for N=8192:
//   out[0:N]          = next_holding(n) = a_n * (1 - prod_m (1 - h_m * D[m,n]))
//   out[N : N+N*N]    = next_domino(r,c) = 1 - (1 - D*(1-a_c))*(1 - D*(1-h_r))
//
// Roofline: read 256 MB (dominos) + write 256 MB (next_domino) ~= 537 MB at
// 23.3 TB/s -> ~23 us floor; ~0.8 GFLOP of VALU is noise. Pure HBM streaming;
// WMMA is structurally irrelevant (the reduction is a product, not a matmul).
// CDNA5 lever: deep GLOBAL_LOAD_ASYNC_TO_LDS_B128 pipeline (ASYNCcnt queue,
// data lands in LDS -> no VGPR pressure per outstanding load), NT temporal
// hints both directions so the one-touch stream doesn't thrash L2.
// ---------------------------------------------------------------------------

#define N_COLORS 8192
#define BLOCK 256
#define COLS_PER_TILE 128               // 32 lanes * 4 floats = 512B/row/wave
#define ROW_CHUNKS 16
#define ROWS_PER_CHUNK (N_COLORS / ROW_CHUNKS)  // 512
#define WAVES 8                          // BLOCK / 32
#define ITERS (ROWS_PER_CHUNK / WAVES)   // 64 rows per thread
#define DEPTH 8                          // async ring depth
#define MAIN_ITERS (ITERS - DEPTH)       // iterations with exactly DEPTH in flight

typedef __attribute__((ext_vector_type(4))) float v4f;

static_assert(DEPTH == 8, "s_wait_asynccnt immediate below assumes DEPTH==8");
static_assert(MAIN_ITERS >= 0, "pipeline deeper than trip count");

__global__ __launch_bounds__(BLOCK)
void wm_elem_partial(const float* __restrict__ action,
                     const float* __restrict__ holding,
                     const float* __restrict__ dominos,
                     float* __restrict__ out,          // d_out base; out2 at +N_COLORS
                     float* __restrict__ ws_partial)   // [ROW_CHUNKS][N_COLORS]
{
    __shared__ v4f stage[DEPTH][BLOCK];   // 32 KB async staging ring
    __shared__ v4f red[BLOCK];            // 4 KB cross-wave product reduction

    const int tid  = threadIdx.x;
    const int lane = tid & 31;
    const int wave = tid >> 5;
    const int colBase = blockIdx.x * COLS_PER_TILE;
    const int col4    = colBase + lane * 4;
    const int rowBase = blockIdx.y * ROWS_PER_CHUNK;

    const v4f av = *(const v4f*)(action + col4);
    const float am0 = 1.0f - av.x, am1 = 1.0f - av.y;
    const float am2 = 1.0f - av.z, am3 = 1.0f - av.w;

    // Generic pointer to LDS: low 32 bits == byte offset within the
    // workgroup's LDS allocation (what the async VDST operand wants).
    const uint32_t lds_base = (uint32_t)(uintptr_t)&stage[0][tid];

    const float* gbase = dominos + (size_t)(rowBase + wave) * N_COLORS + col4;
    const size_t rowStride = (size_t)WAVES * N_COLORS;  // elements per pipeline step

    // ---- prologue: fill the async ring -----------------------------------
#pragma unroll
    for (int i = 0; i < DEPTH; ++i) {
        const uint32_t lds_off = lds_base + (uint32_t)(i * BLOCK * sizeof(v4f));
        const float* gp = gbase + (size_t)i * rowStride;
        asm volatile("global_load_async_to_lds_b128 %0, %1, off th:TH_LOAD_NT"
                     :: "v"(lds_off), "v"(gp) : "memory");
    }

    float p0 = 1.0f, p1 = 1.0f, p2 = 1.0f, p3 = 1.0f;

    // ---- main loop: exactly DEPTH loads in flight at each wait ------------
    for (int i = 0; i < MAIN_ITERS; ++i) {
        const int slot = i & (DEPTH - 1);

        // 8 outstanding -> wait <=7 guarantees the oldest (load i) landed
        asm volatile("s_wait_asynccnt 7" ::: "memory");
        const v4f d = stage[slot][tid];

        const int   row = rowBase + wave + i * WAVES;
        const float h   = holding[row];
        const float hm  = 1.0f - h;

        // next_domino = 1 - (1 - d*(1-a)) * (1 - d*(1-h))
        v4f q;
        q.x = fmaf(-fmaf(-d.x, am0, 1.0f), fmaf(-d.x, hm, 1.0f), 1.0f);
        q.y = fmaf(-fmaf(-d.y, am1, 1.0f), fmaf(-d.y, hm, 1.0f), 1.0f);
        q.z = fmaf(-fmaf(-d.z, am2, 1.0f), fmaf(-d.z, hm, 1.0f), 1.0f);
        q.w = fmaf(-fmaf(-d.w, am3, 1.0f), fmaf(-d.w, hm, 1.0f), 1.0f);
        v4f* outp = (v4f*)(out + (size_t)N_COLORS + (size_t)row * N_COLORS + col4);
        __builtin_nontemporal_store(q, outp);   // one-touch stream: keep L2 clean

        // column partial products: prod *= (1 - h * d)
        p0 *= fmaf(-h, d.x, 1.0f);
        p1 *= fmaf(-h, d.y, 1.0f);
        p2 *= fmaf(-h, d.z, 1.0f);
        p3 *= fmaf(-h, d.w, 1.0f);

        // WAR guard: ds_load of this slot must retire before the async write
        // reuses it (async LDS writes are unordered vs DS ops). DScnt is
        // already 0 here in practice (compiler waited before the FMAs).
        asm volatile("s_wait_dscnt 0x0" ::: "memory");
        const uint32_t lds_off = lds_base + (uint32_t)(slot * BLOCK * sizeof(v4f));
        const float* gp = gbase + (size_t)(i + DEPTH) * rowStride;
        asm volatile("global_load_async_to_lds_b128 %0, %1, off th:TH_LOAD_NT"
                     :: "v"(lds_off), "v"(gp) : "memory");
    }

    // ---- tail: <DEPTH outstanding, so drain fully then consume ------------
    asm volatile("s_wait_asynccnt 0" ::: "memory");
#pragma unroll
    for (int i = MAIN_ITERS; i < ITERS; ++i) {
        const int slot = i & (DEPTH - 1);
        const v4f d = stage[slot][tid];

        const int   row = rowBase + wave + i * WAVES;
        const float h   = holding[row];
        const float hm  = 1.0f - h;

        v4f q;
        q.x = fmaf(-fmaf(-d.x, am0, 1.0f), fmaf(-d.x, hm, 1.0f), 1.0f);
        q.y = fmaf(-fmaf(-d.y, am1, 1.0f), fmaf(-d.y, hm, 1.0f), 1.0f);
        q.z = fmaf(-fmaf(-d.z, am2, 1.0f), fmaf(-d.z, hm, 1.0f), 1.0f);
        q.w = fmaf(-fmaf(-d.w, am3, 1.0f), fmaf(-d.w, hm, 1.0f), 1.0f);
        v4f* outp = (v4f*)(out + (size_t)N_COLORS + (size_t)row * N_COLORS + col4);
        __builtin_nontemporal_store(q, outp);

        p0 *= fmaf(-h, d.x, 1.0f);
        p1 *= fmaf(-h, d.y, 1.0f);
        p2 *= fmaf(-h, d.z, 1.0f);
        p3 *= fmaf(-h, d.w, 1.0f);
    }

    // ---- reduce partial products across the 8 waves ----------------------
    v4f mine; mine.x = p0; mine.y = p1; mine.z = p2; mine.w = p3;
    red[tid] = mine;
    __syncthreads();
    if (tid < 32) {
        v4f r = red[tid];
#pragma unroll
        for (int w = 1; w < WAVES; ++w) {
            const v4f s = red[w * 32 + tid];
            r.x *= s.x; r.y *= s.y; r.z *= s.z; r.w *= s.w;
        }
        *(v4f*)(ws_partial + (size_t)blockIdx.y * N_COLORS + colBase + tid * 4) = r;
    }
}

__global__ __launch_bounds__(256)
void wm_holding(const float* __restrict__ action,
                const float* __restrict__ ws_partial,
                float* __restrict__ out)
{
    const int n = blockIdx.x * blockDim.x + threadIdx.x;
    float prod = 1.0f;
#pragma unroll
    for (int rc = 0; rc < ROW_CHUNKS; ++rc)
        prod *= ws_partial[(size_t)rc * N_COLORS + n];
    out[n] = action[n] * (1.0f - prod);
}

extern "C" void kernel_launch(void* const* d_in, const int* in_sizes, int n_in,
                              void* d_out, int out_size, void* d_ws, size_t ws_size,
                              hipStream_t stream) {
    const float* action  = (const float*)d_in[0];   // [8192]
    const float* holding = (const float*)d_in[1];   // [8192]
    const float* dominos = (const float*)d_in[2];   // [8192 * 8192]
    float* out = (float*)d_out;                     // [8192 + 8192*8192]
    float* ws  = (float*)d_ws;                      // needs 16*8192*4 = 512 KB

    dim3 grid1(N_COLORS / COLS_PER_TILE, ROW_CHUNKS);   // 64 x 16 = 1024 blocks
    wm_elem_partial<<<grid1, BLOCK, 0, stream>>>(action, holding, dominos, out, ws);

    wm_holding<<<N_COLORS / 256, 256, 0, stream>>>(action, ws, out);
}